// CRF_60653528154688
// MI455X (gfx1250) — compile-verified
//
#include <hip/hip_runtime.h>
#include <math.h>

#define Bn 1024
#define Tn 512
#define Kn 64
#define START_TAG 62
#define STOP_TAG 63
#define NEGV (-10000.0f)

typedef float v2f __attribute__((ext_vector_type(2)));
typedef float v8f __attribute__((ext_vector_type(8)));

// ---------------------------------------------------------------------------
// Kernel 2: CRF forward algorithm. One wave (32 threads) per 16-batch tile.
// alpha kept in WMMA C-fragment layout; inner K x K logsumexp lowered to
// v_wmma_f32_16x16x4_f32 on E = exp(trans) after per-batch max subtraction.
// ---------------------------------------------------------------------------
__global__ __launch_bounds__(32) void crf_forward_wmma(
    const float* __restrict__ feats,    // [B,T,K]
    const int*   __restrict__ lengths,  // [B]
    const float* __restrict__ trans,    // [K,K]
    float*       __restrict__ fwdOut)   // [B]
{
    __shared__ float Elds[Kn * Kn];   // exp(trans), row-major [i][j]
    __shared__ float aExp[Kn * 16];   // exp(alpha - m), [tag][batch]
    __shared__ float tStop[Kn];       // trans[STOP][i]

    const int lane  = threadIdx.x;    // 0..31
    const int b     = lane & 15;      // batch within tile
    const int hi    = lane >> 4;      // half-wave selector
    const int bBase = blockIdx.x * 16;

    // ---- init E = exp(trans) and the STOP transition row ----
    for (int idx = lane; idx < Kn * Kn; idx += 32)
        Elds[idx] = expf(trans[idx]);            // exp(-10000) -> 0
    for (int idx = lane; idx < Kn; idx += 32)
        tStop[idx] = trans[STOP_TAG * Kn + idx];
    __syncthreads();

    // ---- alpha in C-fragment layout: tag i = mt*16 + hi*8 + r ----
    float alphaR[4][8];
#pragma unroll
    for (int mt = 0; mt < 4; ++mt)
#pragma unroll
        for (int r = 0; r < 8; ++r) {
            const int i = mt * 16 + hi * 8 + r;
            alphaR[mt][r] = (i == START_TAG) ? 0.0f : NEGV;
        }

    const int myLen = lengths[bBase + b];
    const float* featB = feats + (size_t)(bBase + b) * Tn * Kn;

    for (int t = 0; t < Tn; ++t) {
        // ---- per-batch max m over all 64 tags (lane pair shares a batch) ----
        float m = -__builtin_inff();
#pragma unroll
        for (int mt = 0; mt < 4; ++mt)
#pragma unroll
            for (int r = 0; r < 8; ++r)
                m = fmaxf(m, alphaR[mt][r]);
        m = fmaxf(m, __shfl_xor(m, 16, 32));

        // ---- a[j] = exp(alpha[j] - m) into LDS [tag][batch] ----
#pragma unroll
        for (int mt = 0; mt < 4; ++mt)
#pragma unroll
            for (int r = 0; r < 8; ++r) {
                const int i = mt * 16 + hi * 8 + r;
                aExp[i * 16 + b] = expf(alphaR[mt][r] - m);
            }
        __syncthreads();

        // ---- gather B fragments: lane holds K = 4c+2*hi, 4c+2*hi+1 ----
        v2f Bf[16];
#pragma unroll
        for (int c = 0; c < 16; ++c) {
            const int kb = 4 * c + 2 * hi;
            Bf[c].x = aExp[(kb + 0) * 16 + b];
            Bf[c].y = aExp[(kb + 1) * 16 + b];
        }

        // ---- D[i, batch] = sum_j E[i,j] * a[j, batch] via WMMA f32 16x16x4 ----
#pragma unroll
        for (int mt = 0; mt < 4; ++mt) {
            v8f acc = {0.f, 0.f, 0.f, 0.f, 0.f, 0.f, 0.f, 0.f};
#pragma unroll
            for (int c = 0; c < 16; ++c) {
                const int kb = 4 * c + 2 * hi;
                // A fragment: row i = mt*16 + (lane&15), K pair at kb
                const v2f Af = *(const v2f*)&Elds[(mt * 16 + b) * Kn + kb];
                acc = __builtin_amdgcn_wmma_f32_16x16x4_f32(
                    false, Af, false, Bf[c], (short)0, acc, false, false);
            }
            // ---- alpha'[i] = emit[b,t,i] + m + log(acc) ----
            const float* ep = featB + (size_t)t * Kn + mt * 16 + hi * 8;
            const float4 e0 = *(const float4*)(ep + 0);
            const float4 e1 = *(const float4*)(ep + 4);
            alphaR[mt][0] = e0.x + m + logf(acc[0]);
            alphaR[mt][1] = e0.y + m + logf(acc[1]);
            alphaR[mt][2] = e0.z + m + logf(acc[2]);
            alphaR[mt][3] = e0.w + m + logf(acc[3]);
            alphaR[mt][4] = e1.x + m + logf(acc[4]);
            alphaR[mt][5] = e1.y + m + logf(acc[5]);
            alphaR[mt][6] = e1.z + m + logf(acc[6]);
            alphaR[mt][7] = e1.w + m + logf(acc[7]);
        }

        // ---- capture terminal logsumexp when this batch finishes ----
        if (t + 1 == myLen) {
            float term[4][8];
            float mx = -__builtin_inff();
#pragma unroll
            for (int mt = 0; mt < 4; ++mt)
#pragma unroll
                for (int r = 0; r < 8; ++r) {
                    const int i = mt * 16 + hi * 8 + r;
                    term[mt][r] = alphaR[mt][r] + tStop[i];
                    mx = fmaxf(mx, term[mt][r]);
                }
            mx = fmaxf(mx, __shfl_xor(mx, 16, 32));
            float s = 0.0f;
#pragma unroll
            for (int mt = 0; mt < 4; ++mt)
#pragma unroll
                for (int r = 0; r < 8; ++r)
                    s += expf(term[mt][r] - mx);
            s += __shfl_xor(s, 16, 32);
            if (hi == 0)
                fwdOut[bBase + b] = mx + logf(s);
        }
        __syncthreads();  // protect aExp reads from next iteration's writes
    }
}

// ---------------------------------------------------------------------------
// Kernel 1: gold path score per batch (trans path score + emission score)
// ---------------------------------------------------------------------------
__global__ __launch_bounds__(64) void crf_gold(
    const float* __restrict__ feats,
    const int*   __restrict__ lengths,
    const int*   __restrict__ tags,
    const float* __restrict__ trans,
    float*       __restrict__ goldOut)
{
    const int b   = blockIdx.x;
    const int tid = threadIdx.x;
    const int len = lengths[b];
    const int*   tg = tags + (size_t)b * Tn;
    const float* fp = feats + (size_t)b * Tn * Kn;

    float s = 0.0f;
    for (int t = tid; t <= len; t += 64) {
        const int ps = (t == 0)   ? START_TAG : tg[t - 1];
        const int pe = (t >= len) ? STOP_TAG  : tg[t];
        s += trans[pe * Kn + ps];
        if (t < len)
            s += fp[(size_t)t * Kn + tg[t]];
    }
    __shared__ float red[64];
    red[tid] = s;
    __syncthreads();
    for (int off = 32; off > 0; off >>= 1) {
        if (tid < off) red[tid] += red[tid + off];
        __syncthreads();
    }
    if (tid == 0) goldOut[b] = red[0];
}

// ---------------------------------------------------------------------------
// Kernel 3: deterministic mean(forward - gold)
// ---------------------------------------------------------------------------
__global__ __launch_bounds__(256) void crf_mean(
    const float* __restrict__ fwd,
    const float* __restrict__ gold,
    float*       __restrict__ out)
{
    __shared__ float red[256];
    const int tid = threadIdx.x;
    float s = 0.0f;
    for (int i = tid; i < Bn; i += 256)
        s += fwd[i] - gold[i];
    red[tid] = s;
    __syncthreads();
    for (int off = 128; off > 0; off >>= 1) {
        if (tid < off) red[tid] += red[tid + off];
        __syncthreads();
    }
    if (tid == 0) out[0] = red[0] * (1.0f / (float)Bn);
}

extern "C" void kernel_launch(void* const* d_in, const int* in_sizes, int n_in,
                              void* d_out, int out_size, void* d_ws, size_t ws_size,
                              hipStream_t stream) {
    const float* feats   = (const float*)d_in[0];   // [B,T,K]
    const int*   lengths = (const int*)  d_in[1];   // [B]
    const int*   tags    = (const int*)  d_in[2];   // [B,T]
    const float* trans   = (const float*)d_in[3];   // [K,K]
    float* out = (float*)d_out;

    float* fwd  = (float*)d_ws;        // [B]
    float* gold = fwd + Bn;            // [B]

    crf_gold<<<Bn, 64, 0, stream>>>(feats, lengths, tags, trans, gold);
    crf_forward_wmma<<<Bn / 16, 32, 0, stream>>>(feats, lengths, trans, fwd);
    crf_mean<<<1, 256, 0, stream>>>(fwd, gold, out);
}